// LocallyConnected2d_73787538145486
// MI455X (gfx1250) — compile-verified
//
#include <hip/hip_runtime.h>

typedef __attribute__((ext_vector_type(2))) float v2f;
typedef __attribute__((ext_vector_type(8))) float v8f;
typedef int v4i_g __attribute__((vector_size(16)));   // matches builtin's int4

#define L_TOT   3136
#define OW_DIM  56
#define CIN     64
#define COUT    64
#define FEAT    576      // 64 * 3 * 3
#define BATCH   32
#define HW      3136     // 56*56
#define KC      144      // K-chunk size (4 chunks of 144)

// LDS strides chosen so stride % 64 has gcd 4 with 64 -> the paired b64
// fragment loads (16 lanes at n*S+k, 16 lanes at n*S+k+2) hit all 64 banks.
#define SA      580      // patches stride   (580 % 64 == 4)
#define SWC     148      // weight-chunk row stride (148 % 64 == 20)

#define LDS_A_FLOATS     (BATCH * SA)       // 18560
#define LDS_WBUF_FLOATS  (COUT * SWC)       // 9472
#define LDS_TOTAL_FLOATS (LDS_A_FLOATS + 2 * LDS_WBUF_FLOATS)   // 37504
#define LDS_BYTES        (LDS_TOTAL_FLOATS * 4)                 // 150016 -> 2 WG/WGP

#if __has_builtin(__builtin_amdgcn_global_load_async_to_lds_b128)
#define HAVE_ASYNC 1
#else
#define HAVE_ASYNC 0
#endif

#if HAVE_ASYNC
#define WAIT_ASYNC(n) __builtin_amdgcn_s_wait_asynccnt(n)   // n must be literal
#else
#define WAIT_ASYNC(n) do { } while (0)
#endif

// Copy one 64 x 144 f32 weight chunk (rows of 576B) into an LDS buffer with
// row stride 148 floats. 4 threads per row, 9 x b128 per thread (2304 segs).
__device__ __forceinline__ void copy_w_chunk(const float* __restrict__ wl,
                                             float* __restrict__ buf,
                                             int t, int tid) {
    const int row = tid >> 2;                 // 0..63
    const int c0  = tid & 3;                  // float4 slot within row
    const float* src = wl + row * FEAT + t * KC + c0 * 4;
    float*       dst = buf + row * SWC + c0 * 4;  // 16B-aligned (592B rows)
#pragma unroll
    for (int it = 0; it < 9; ++it) {
#if HAVE_ASYNC
        __builtin_amdgcn_global_load_async_to_lds_b128(
            (__attribute__((address_space(1))) v4i_g*)(src + it * 16),
            (__attribute__((address_space(3))) v4i_g*)(dst + it * 16),
            /*offset=*/0, /*cpol=*/0);
#else
        float4 v = *(const float4*)(src + it * 16);
        *(float4*)(dst + it * 16) = v;
#endif
    }
}

__device__ __forceinline__ void compute_chunk(const float* __restrict__ pa,
                                              const float* __restrict__ pw,
                                              v8f& acc) {
#pragma unroll 8
    for (int k = 0; k < KC; k += 4) {
        v2f a = *(const v2f*)(pa + k);
        v2f b = *(const v2f*)(pw + k);
        acc = __builtin_amdgcn_wmma_f32_16x16x4_f32(
            /*neg_a=*/false, a, /*neg_b=*/false, b,
            /*c_mod=*/(short)0, acc, /*reuse_a=*/false, /*reuse_b=*/false);
    }
}

__global__ __launch_bounds__(256) void lc2d_wmma_async_kernel(
    const float* __restrict__ x,      // [B, CIN, 56, 56]
    const float* __restrict__ w,      // [L, COUT, FEAT]
    const float* __restrict__ bias,   // [L, COUT]
    float* __restrict__ out)          // [B, COUT, L]
{
    extern __shared__ __attribute__((aligned(16))) float smem[];
    float* ldsA  = smem;                                  // [BATCH][SA]
    float* ldsW0 = smem + LDS_A_FLOATS;                   // [COUT][SWC] buf 0
    float* ldsW1 = ldsW0 + LDS_WBUF_FLOATS;               // [COUT][SWC] buf 1

    const int tid = threadIdx.x;
    const int l   = blockIdx.x;
    const int oh  = l / OW_DIM;
    const int ow  = l % OW_DIM;
    const float* wl = w + (size_t)l * (COUT * FEAT);

    // Kick off the HBM weight stream first: chunks 0 and 1 (async -> LDS).
    copy_w_chunk(wl, ldsW0, 0, tid);
    copy_w_chunk(wl, ldsW1, 1, tid);

    // im2col gather of patches (L2-resident x) overlaps the async stream.
    for (int p = tid; p < BATCH * CIN; p += 256) {        // 8 (b,c) per thread
        int b = p >> 6;
        int c = p & 63;
        const float* xp = x + ((size_t)(b * CIN + c)) * HW;
        float* dst = ldsA + b * SA + c * 9;
#pragma unroll
        for (int ki = 0; ki < 3; ++ki) {
            int ih = oh - 1 + ki;
            bool okh = (unsigned)ih < (unsigned)OW_DIM;
#pragma unroll
            for (int kj = 0; kj < 3; ++kj) {
                int iw = ow - 1 + kj;
                bool ok = okh && ((unsigned)iw < (unsigned)OW_DIM);
                dst[ki * 3 + kj] = ok ? xp[ih * OW_DIM + iw] : 0.0f;
            }
        }
    }

    // Wave -> output tile mapping.
    const int wave = tid >> 5;
    const int lane = tid & 31;
    const int mi = wave & 1;        // batch tile  (0..1)
    const int ni = wave >> 1;       // cout  tile  (0..3)
    const int lr = lane & 15;
    const int hh = lane >> 4;

    const int   cg = ni * 16 + lr;
    const float bv = bias[(size_t)l * COUT + cg];         // hoisted early

    const float* pa0 = ldsA + (mi * 16 + lr) * SA + 2 * hh;
    const float* pw0 = ldsW0 + (ni * 16 + lr) * SWC + 2 * hh;
    const float* pw1 = ldsW1 + (ni * 16 + lr) * SWC + 2 * hh;

    v8f acc = {0.f, 0.f, 0.f, 0.f, 0.f, 0.f, 0.f, 0.f};

    // ---- Manually unrolled 4-stage software pipeline (literal wait counts) --
    WAIT_ASYNC(9);              // chunk 0 arrived (9 newest = chunk 1 in flight)
    __syncthreads();

    compute_chunk(pa0 + 0 * KC, pw0, acc);      // chunk 0 (buf 0)
    __syncthreads();                            // buf 0 free
    copy_w_chunk(wl, ldsW0, 2, tid);            // chunk 2 -> buf 0 (async)
    WAIT_ASYNC(9);                              // chunk 1 fully arrived
    __syncthreads();

    compute_chunk(pa0 + 1 * KC, pw1, acc);      // chunk 1 (buf 1)
    __syncthreads();                            // buf 1 free
    copy_w_chunk(wl, ldsW1, 3, tid);            // chunk 3 -> buf 1 (async)
    WAIT_ASYNC(9);                              // chunk 2 fully arrived
    __syncthreads();

    compute_chunk(pa0 + 2 * KC, pw0, acc);      // chunk 2 (buf 0)
    WAIT_ASYNC(0);                              // chunk 3 fully arrived
    __syncthreads();

    compute_chunk(pa0 + 3 * KC, pw1, acc);      // chunk 3 (buf 1)

    // Epilogue: D layout VGPR j -> M = j + 8*half, N = lane&15.
#pragma unroll
    for (int j = 0; j < 8; ++j) {
        int bg = mi * 16 + hh * 8 + j;
        out[((size_t)(bg * COUT + cg)) * L_TOT + l] = acc[j] + bv;
    }
}

extern "C" void kernel_launch(void* const* d_in, const int* in_sizes, int n_in,
                              void* d_out, int out_size, void* d_ws, size_t ws_size,
                              hipStream_t stream) {
    const float* x    = (const float*)d_in[0];   // 32*64*56*56
    const float* w    = (const float*)d_in[1];   // 3136*64*576
    const float* bias = (const float*)d_in[2];   // 3136*64
    float* out = (float*)d_out;                  // 32*64*3136

    (void)hipFuncSetAttribute((const void*)lc2d_wmma_async_kernel,
                              hipFuncAttributeMaxDynamicSharedMemorySize, LDS_BYTES);

    dim3 grid(L_TOT);
    dim3 block(256);
    lc2d_wmma_async_kernel<<<grid, block, LDS_BYTES, stream>>>(x, w, bias, out);
}